// MMD_loss_59614146069008
// MI455X (gfx1250) — compile-verified
//
#include <hip/hip_runtime.h>
#include <math.h>

#define HD 256                  // hidden dim (K of the Gram GEMM)
#define LDS_PAD 4               // bank-conflict pad: stride%64==4 spreads 4*lane+2*half over all 64 banks
#define ROWSTR (HD + LDS_PAD)   // 260 floats per staged row

typedef __attribute__((ext_vector_type(2))) float v2f;  // A/B frag of V_WMMA_F32_16X16X4_F32
typedef __attribute__((ext_vector_type(8))) float v8f;  // 16x16 f32 C/D frag

// ---- workspace layout (float indices) ----
// f[0]=ns  f[1]=nt  f[2]=bw_num  f[3]=bandwidth  f[4]=result
// f[8..23]=log2e/(bw*mul^t)  f[24]=kernel_num  f[25]=1/ns^2 f[26]=1/nt^2 f[27]=1/(ns*nt)
// f[32 .. 32+N)        = sq norms
// f[32+N .. 32+2N)     = weights (mask as float)

__global__ void init_ws(float* __restrict__ f) {
  if (threadIdx.x < 32) f[threadIdx.x] = 0.0f;
}

__global__ __launch_bounds__(256) void rownorm(
    const float* __restrict__ src, const float* __restrict__ tgt,
    const int* __restrict__ smask, const int* __restrict__ tmask,
    float* __restrict__ f, int N1, int N) {
  int wave = threadIdx.x >> 5;
  int lane = threadIdx.x & 31;
  int r = blockIdx.x * 8 + wave;
  if (r >= N) return;
  const float* row = (r < N1) ? (src + (size_t)r * HD)
                              : (tgt + (size_t)(r - N1) * HD);
  const float4* row4 = (const float4*)row;     // 1KB-aligned rows
  float4 a = row4[lane * 2 + 0];
  float4 b = row4[lane * 2 + 1];
  float s = a.x*a.x + a.y*a.y + a.z*a.z + a.w*a.w
          + b.x*b.x + b.y*b.y + b.z*b.z + b.w*b.w;
  for (int off = 16; off; off >>= 1) s += __shfl_xor(s, off, 32);
  if (lane == 0) {
    f[32 + r] = s;
    float w = (float)((r < N1) ? smask[r] : tmask[r - N1]);
    f[32 + N + r] = w;
    atomicAdd((r < N1) ? &f[0] : &f[1], w);
  }
}

// One wave = one 16x16 tile via V_WMMA_F32_16X16X4_F32; block = 8 waves = 32x64 tile.
// MODE 0: accumulate sum_ij w_i w_j D_ij into f[2]
// MODE 1: accumulate sum_ij coef_ij * w_i w_j * sum_t exp2(-D_ij*scale_t) into f[4]
template <int MODE>
__global__ __launch_bounds__(256) void tile_kernel(
    const float* __restrict__ src, const float* __restrict__ tgt,
    float* __restrict__ f, int N1, int N) {
  extern __shared__ float smem[];
  float* As  = smem;                  // 32 rows * ROWSTR
  float* Bs  = smem + 32 * ROWSTR;    // 64 rows * ROWSTR
  float* red = smem + 96 * ROWSTR;    // 8 floats

  const float* sq = f + 32;
  const float* wb = f + 32 + N;

  const int row0 = blockIdx.y * 32;
  const int col0 = blockIdx.x * 64;
  const int t = threadIdx.x;

  // ---- stage A band (32 rows x 256 f32) with coalesced b128 loads ----
  for (int chunk = 0; chunk < 8; ++chunk) {
    int idx = chunk * 256 + t;        // 0..2047 float4 slots
    int r = idx >> 6, c = idx & 63;
    int gr = row0 + r; if (gr >= N) gr = N - 1;
    const float* rowp = (gr < N1) ? (src + (size_t)gr * HD)
                                  : (tgt + (size_t)(gr - N1) * HD);
    *(float4*)&As[r * ROWSTR + c * 4] = ((const float4*)rowp)[c];
  }
  // ---- stage B band (64 rows x 256 f32) ----
  for (int chunk = 0; chunk < 16; ++chunk) {
    int idx = chunk * 256 + t;
    int r = idx >> 6, c = idx & 63;
    int gc = col0 + r; if (gc >= N) gc = N - 1;
    const float* rowp = (gc < N1) ? (src + (size_t)gc * HD)
                                  : (tgt + (size_t)(gc - N1) * HD);
    *(float4*)&Bs[r * ROWSTR + c * 4] = ((const float4*)rowp)[c];
  }
  __syncthreads();

  const int wave = t >> 5, lane = t & 31;
  const int wrow = (wave >> 2) * 16;          // 0 / 16
  const int wcol = (wave & 3) * 16;           // 0 / 16 / 32 / 48
  const int half = lane >> 4, l = lane & 15;

  // A frag: lane l holds row M=l, K = 2*half + {0,1}; B identical with N=l.
  const float* ap = &As[(wrow + l) * ROWSTR + 2 * half];
  const float* bp = &Bs[(wcol + l) * ROWSTR + 2 * half];

  v8f acc0 = {};
  v8f acc1 = {};
#pragma unroll 8
  for (int k = 0; k < HD; k += 8) {
    v2f a0 = *(const v2f*)(ap + k);
    v2f b0 = *(const v2f*)(bp + k);
    acc0 = __builtin_amdgcn_wmma_f32_16x16x4_f32(false, a0, false, b0,
                                                 (short)0, acc0, false, false);
    v2f a1 = *(const v2f*)(ap + k + 4);
    v2f b1 = *(const v2f*)(bp + k + 4);
    acc1 = __builtin_amdgcn_wmma_f32_16x16x4_f32(false, a1, false, b1,
                                                 (short)0, acc1, false, false);
  }

  // hoist pass-2 constants
  float scl[16]; int kn = 0; float cSS = 0.f, cTT = 0.f, cST = 0.f;
  if (MODE == 1) {
    kn = (int)f[24]; if (kn > 16) kn = 16;
    for (int i = 0; i < 16; ++i) scl[i] = f[8 + i];
    cSS = f[25]; cTT = f[26]; cST = f[27];
  }

  // C/D layout: acc[v] -> row i = row0+wrow + (half ? v+8 : v), col j = col0+wcol+l
  const int j = col0 + wcol + l;
  const float wj = wb[j];
  const float sqj = sq[j];
  float part = 0.0f;
#pragma unroll
  for (int v = 0; v < 8; ++v) {
    const int i = row0 + wrow + (half ? v + 8 : v);
    const float g = acc0[v] + acc1[v];
    float d = sq[i] + sqj - 2.0f * g;
    d = fmaxf(d, 0.0f);
    const float ww = wb[i] * wj;
    if (MODE == 0) {
      part += ww * d;
    } else {
      float ks = 0.0f;
      for (int tt = 0; tt < kn; ++tt) ks += exp2f(-d * scl[tt]);  // v_exp_f32
      const float coef = (i < N1) ? ((j < N1) ? cSS : -cST)
                                  : ((j < N1) ? -cST : cTT);
      part += coef * ww * ks;
    }
  }
  for (int off = 16; off; off >>= 1) part += __shfl_xor(part, off, 32);
  if (lane == 0) red[wave] = part;
  __syncthreads();
  if (t == 0) {
    float s = 0.0f;
    for (int w = 0; w < 8; ++w) s += red[w];
    atomicAdd(&f[(MODE == 0) ? 2 : 4], s);
  }
}

__global__ void bw_finalize(float* __restrict__ f,
                            const int* __restrict__ kmul_p,
                            const int* __restrict__ knum_p) {
  const float ns = f[0], nt = f[1];
  const float n = ns + nt;
  int kmul = *kmul_p;
  int knum = *knum_p; if (knum > 16) knum = 16;
  float bw = f[2] / (n * n - n);
  float div = 1.0f;
  for (int i = 0; i < knum / 2; ++i) div *= (float)kmul;  // kernel_mul^(kernel_num//2)
  bw /= div;
  f[3] = bw;
  const float LOG2E = 1.4426950408889634f;
  float inv = 1.0f / bw;
  for (int i = 0; i < 16; ++i) {
    f[8 + i] = (i < knum) ? (LOG2E * inv) : 0.0f;
    if (i < knum) inv /= (float)kmul;
  }
  f[24] = (float)knum;
  f[25] = 1.0f / (ns * ns);
  f[26] = 1.0f / (nt * nt);
  f[27] = 1.0f / (ns * nt);
}

__global__ void writeout(const float* __restrict__ f, float* __restrict__ out) {
  if (threadIdx.x == 0 && blockIdx.x == 0) out[0] = f[4];
}

extern "C" void kernel_launch(void* const* d_in, const int* in_sizes, int n_in,
                              void* d_out, int out_size, void* d_ws, size_t ws_size,
                              hipStream_t stream) {
  const float* src  = (const float*)d_in[0];
  const float* tgt  = (const float*)d_in[1];
  const int* smask  = (const int*)d_in[2];
  const int* tmask  = (const int*)d_in[3];
  const int* kmul   = (const int*)d_in[4];
  const int* knum   = (const int*)d_in[5];
  const int N1 = in_sizes[0] / HD;
  const int N2 = in_sizes[1] / HD;
  const int N  = N1 + N2;
  float* f = (float*)d_ws;

  init_ws<<<1, 32, 0, stream>>>(f);
  rownorm<<<(N + 7) / 8, 256, 0, stream>>>(src, tgt, smask, tmask, f, N1, N);

  dim3 grid((N + 63) / 64, (N + 31) / 32);
  const size_t shmem = (size_t)(96 * ROWSTR + 8) * sizeof(float);  // ~97.6 KB of 320 KB/WGP
  tile_kernel<0><<<grid, 256, shmem, stream>>>(src, tgt, f, N1, N);
  bw_finalize<<<1, 1, 0, stream>>>(f, kmul, knum);
  tile_kernel<1><<<grid, 256, shmem, stream>>>(src, tgt, f, N1, N);
  writeout<<<1, 1, 0, stream>>>(f, (float*)d_out);
}